// UncertaintyHead_91061896609905
// MI455X (gfx1250) — compile-verified
//
#include <hip/hip_runtime.h>
#include <hip/hip_bf16.h>

// Problem sizes (from reference): N=4096, D=1024, C=1000, T=32
#define PN 4096
#define PD 1024
#define PC 1000
#define PT 32
#define NCOLS (2 * PC)            // 2000 columns of h
#define TOTAL_EPS 131072000u      // T*N*C
#define HALF_EPS  65536000u       // TOTAL_EPS / 2 == 16*N*C

typedef __attribute__((ext_vector_type(2))) float v2f;
typedef __attribute__((ext_vector_type(8))) float v8f;

// ---------------------------------------------------------------------------
// Kernel 1: h = x @ W^T  via V_WMMA_F32_16X16X4_F32 (exact fp32 matrix path)
//
// Per-wave tile: 16 (M) x 80 (N) = 5 accumulators of v8f, K swept in steps of 4.
// A 16x4 layout (ISA 7.12.2): lanes 0-15 -> M, VGPR r holds K = 2*(lane>=16)+r.
// B  4x16 layout (mirror):    lanes 0-15 -> N, VGPR r holds K = 2*(lane>=16)+r.
// All 8 waves of a block share the same 80-column B panel -> WGP$ hits.
// ---------------------------------------------------------------------------
__global__ __launch_bounds__(256) void gemm_h_kernel(const float* __restrict__ x,
                                                     const float* __restrict__ W,
                                                     float* __restrict__ h) {
    const int lane = threadIdx.x & 31;
    const int wave = threadIdx.x >> 5;
    const int l16  = lane & 15;
    const int hf   = lane >> 4;                   // 0: lanes 0-15, 1: lanes 16-31

    const int mbase = (blockIdx.x * 8 + wave) * 16;   // 32 blocks * 8 waves = 256 M-tiles
    const int cbase = blockIdx.y * 80;                // 25 blocks * 80 = 2000 cols

    const float* arow = x + (size_t)(mbase + l16) * PD + hf * 2;
    const float* brow = W + (size_t)(cbase + l16) * PD + hf * 2;

    v8f acc[5];
#pragma unroll
    for (int j = 0; j < 5; ++j) acc[j] = (v8f){0.f, 0.f, 0.f, 0.f, 0.f, 0.f, 0.f, 0.f};

#pragma unroll 4
    for (int k = 0; k < PD; k += 4) {
        v2f a = *(const v2f*)(arow + k);
#pragma unroll
        for (int j = 0; j < 5; ++j) {
            v2f b = *(const v2f*)(brow + (size_t)j * 16 * PD + k);
            // 8 args: (neg_a, A, neg_b, B, c_mod, C, reuse_a, reuse_b)
            acc[j] = __builtin_amdgcn_wmma_f32_16x16x4_f32(
                false, a, false, b, (short)0, acc[j], false, false);
        }
    }

    // C/D layout: VGPR r, lanes 0-15 -> (M = r,     N = l16)
    //                     lanes 16-31-> (M = r + 8, N = l16)
#pragma unroll
    for (int j = 0; j < 5; ++j) {
        float* cp = h + (size_t)(mbase + hf * 8) * NCOLS + (cbase + j * 16 + l16);
#pragma unroll
        for (int r = 0; r < 8; ++r) cp[(size_t)r * NCOLS] = acc[j][r];
    }
}

// ---------------------------------------------------------------------------
// JAX-compatible Threefry-2x32 for jax.random.normal(key(42), (T,N,C), f32).
// key data = (0, 42). Counter split: block i lanes are (i, i+HALF), outputs
// bits[i] (x0) and bits[i+HALF] (x1). Since HALF = 16*N*C, element (t,n,c)
// with t<16 pairs with (t+16,n,c): one Threefry yields BOTH eps values.
// normal = sqrt(2) * erfinv(uniform(lo=nextafter(-1,0), hi=1))
// ---------------------------------------------------------------------------
__device__ __forceinline__ unsigned rotl32(unsigned x, int d) {
    return (x << d) | (x >> (32 - d));
}

__device__ __forceinline__ float bits_to_normal(unsigned bits) {
    // uniform in [lo, 1): mantissa bits -> [1,2) -> [0,1) -> scale to [lo,1)
    float f = __uint_as_float((bits >> 9) | 0x3f800000u) - 1.0f;
    const float lo = -0.99999994f;              // nextafter(-1, 0)
    float u = fmaf(f, 1.0f - lo, lo);
    u = fmaxf(u, lo);

    // Giles (2010) single-precision erfinv
    float w = -__logf((1.0f - u) * (1.0f + u));
    float p;
    if (w < 5.0f) {
        w -= 2.5f;
        p = 2.81022636e-08f;
        p = fmaf(p, w, 3.43273939e-07f);
        p = fmaf(p, w, -3.5233877e-06f);
        p = fmaf(p, w, -4.39150654e-06f);
        p = fmaf(p, w, 0.00021858087f);
        p = fmaf(p, w, -0.00125372503f);
        p = fmaf(p, w, -0.00417768164f);
        p = fmaf(p, w, 0.246640727f);
        p = fmaf(p, w, 1.50140941f);
    } else {
        w = __fsqrt_rn(w) - 3.0f;
        p = -0.000200214257f;
        p = fmaf(p, w, 0.000100950558f);
        p = fmaf(p, w, 0.00134934322f);
        p = fmaf(p, w, -0.00367342844f);
        p = fmaf(p, w, 0.00573950773f);
        p = fmaf(p, w, -0.0076224613f);
        p = fmaf(p, w, 0.00943887047f);
        p = fmaf(p, w, 1.00167406f);
        p = fmaf(p, w, 2.83297682f);
    }
    return 1.41421356237f * p * u;
}

// i must be < HALF_EPS. Produces eps for flat indices i (lo) and i+HALF (hi).
__device__ __forceinline__ void jax_normal_pair(unsigned i, float& e_lo, float& e_hi) {
    unsigned x0 = i, x1 = i + HALF_EPS;
    const unsigned k0 = 0u, k1 = 42u;
    const unsigned k2 = k0 ^ k1 ^ 0x1BD11BDAu;
    x0 += k0; x1 += k1;
#define TF_R(r) { x0 += x1; x1 = rotl32(x1, r); x1 ^= x0; }
    TF_R(13) TF_R(15) TF_R(26) TF_R(6)   x0 += k1; x1 += k2 + 1u;
    TF_R(17) TF_R(29) TF_R(16) TF_R(24)  x0 += k2; x1 += k0 + 2u;
    TF_R(13) TF_R(15) TF_R(26) TF_R(6)   x0 += k0; x1 += k1 + 3u;
    TF_R(17) TF_R(29) TF_R(16) TF_R(24)  x0 += k1; x1 += k2 + 4u;
    TF_R(13) TF_R(15) TF_R(26) TF_R(6)   x0 += k2; x1 += k0 + 5u;
#undef TF_R
    e_lo = bits_to_normal(x0);
    e_hi = bits_to_normal(x1);
}

// ---------------------------------------------------------------------------
// Kernel 2: one block per row n. Each iteration of the t-loop handles the
// (t, t+16) pair from one Threefry call: two softmaxes per iteration, their
// cross-wave reductions sharing barriers. 4 columns per thread in registers.
// ---------------------------------------------------------------------------
__global__ __launch_bounds__(256) void sample_softmax_kernel(const float* __restrict__ h,
                                                             float* __restrict__ out) {
    const int n    = blockIdx.x;
    const int tid  = threadIdx.x;
    const int lane = tid & 31;
    const int wid  = tid >> 5;

    __shared__ float s_red[16];   // [0..7] sub-step lo, [8..15] sub-step hi

    float mu[4], sig[4];
    int   cidx[4];
    bool  act[4];
#pragma unroll
    for (int j = 0; j < 4; ++j) {
        const int c = tid + j * 256;
        cidx[j] = c;
        act[j]  = (c < PC);
        if (act[j]) {
            mu[j] = h[(size_t)n * NCOLS + c];
            const float lv = h[(size_t)n * NCOLS + PC + c];
            sig[j] = __expf(0.5f * lv);
            out[(size_t)PN * PC + (size_t)n * PC + c] = sig[j];  // sigma output
        } else {
            mu[j] = 0.0f; sig[j] = 0.0f;
        }
    }

    float acc[4] = {0.f, 0.f, 0.f, 0.f};
    const float NEG = -3.402823466e+38f;

    for (int t = 0; t < PT / 2; ++t) {          // handles t and t+16 together
        const unsigned base = (unsigned)t * (unsigned)(PN * PC) + (unsigned)n * (unsigned)PC;
        float lg0[4], lg1[4];
        float m0 = NEG, m1 = NEG;
#pragma unroll
        for (int j = 0; j < 4; ++j) {
            if (act[j]) {
                float e0, e1;
                jax_normal_pair(base + (unsigned)cidx[j], e0, e1);
                lg0[j] = fmaf(sig[j], e0, mu[j]);
                lg1[j] = fmaf(sig[j], e1, mu[j]);
                m0 = fmaxf(m0, lg0[j]);
                m1 = fmaxf(m1, lg1[j]);
            } else {
                lg0[j] = NEG; lg1[j] = NEG;
            }
        }
        // max reductions (both sub-steps share barriers)
#pragma unroll
        for (int off = 16; off > 0; off >>= 1) {
            m0 = fmaxf(m0, __shfl_xor(m0, off, 32));
            m1 = fmaxf(m1, __shfl_xor(m1, off, 32));
        }
        if (lane == 0) { s_red[wid] = m0; s_red[8 + wid] = m1; }
        __syncthreads();
        float mm0 = s_red[0], mm1 = s_red[8];
#pragma unroll
        for (int w = 1; w < 8; ++w) {
            mm0 = fmaxf(mm0, s_red[w]);
            mm1 = fmaxf(mm1, s_red[8 + w]);
        }
        __syncthreads();

        float e0[4], e1[4];
        float s0 = 0.0f, s1 = 0.0f;
#pragma unroll
        for (int j = 0; j < 4; ++j) {
            e0[j] = act[j] ? __expf(lg0[j] - mm0) : 0.0f;
            e1[j] = act[j] ? __expf(lg1[j] - mm1) : 0.0f;
            s0 += e0[j];
            s1 += e1[j];
        }
#pragma unroll
        for (int off = 16; off > 0; off >>= 1) {
            s0 += __shfl_xor(s0, off, 32);
            s1 += __shfl_xor(s1, off, 32);
        }
        if (lane == 0) { s_red[wid] = s0; s_red[8 + wid] = s1; }
        __syncthreads();
        float ss0 = 0.0f, ss1 = 0.0f;
#pragma unroll
        for (int w = 0; w < 8; ++w) {
            ss0 += s_red[w];
            ss1 += s_red[8 + w];
        }
        __syncthreads();

        const float inv0 = 1.0f / ss0;
        const float inv1 = 1.0f / ss1;
#pragma unroll
        for (int j = 0; j < 4; ++j) {
            acc[j] += e0[j] * inv0 + e1[j] * inv1;
        }
    }

#pragma unroll
    for (int j = 0; j < 4; ++j) {
        if (act[j]) {
            out[(size_t)n * PC + cidx[j]] = __expf(acc[j] * (1.0f / (float)PT));
        }
    }
}

extern "C" void kernel_launch(void* const* d_in, const int* in_sizes, int n_in,
                              void* d_out, int out_size, void* d_ws, size_t ws_size,
                              hipStream_t stream) {
    (void)in_sizes; (void)n_in; (void)out_size; (void)ws_size;
    const float* x = (const float*)d_in[0];   // (N, D)
    const float* W = (const float*)d_in[1];   // (2C, D)
    float* out = (float*)d_out;               // [mu_out (N*C) | sigma (N*C)]
    float* h   = (float*)d_ws;                // (N, 2C) fp32 = 32.77 MB scratch

    dim3 g1(PN / (16 * 8), NCOLS / 80);       // 32 x 25 blocks, 256 threads (8 waves)
    gemm_h_kernel<<<g1, 256, 0, stream>>>(x, W, h);

    sample_softmax_kernel<<<PN, 256, 0, stream>>>(h, out);
}